// DocREModel_12025908428891
// MI455X (gfx1250) — compile-verified
//
#include <hip/hip_runtime.h>
#include <math.h>

// Problem dims (fixed by the reference)
#define B_   4
#define L_   1024
#define H_   768
#define NH_  12
#define E_   32
#define MPE_ 4
#define K_   16
#define EM_  128          // E_*MPE_
#define TD_  20
#define NODES_   176      // E + EM + K
#define CTXROWS_ 160      // E + EM
#define NROWS_   336      // NODES_ + CTXROWS_
#define FDIM_    788      // H + TD

typedef __attribute__((ext_vector_type(2))) float v2f;
typedef __attribute__((ext_vector_type(4))) float v4f;
typedef __attribute__((ext_vector_type(8))) float v8f;

// ---------------------------------------------------------------------------
// K1: att_h[b,r,c] = mean over NH heads of attention[b,h,r,c]
// Dominant stream: 201 MB read once -> non-temporal b128 loads so the
// reused tensors (att_h, seq) stay resident in the 192 MB L2.
// ---------------------------------------------------------------------------
__global__ void k_att_head_mean(const float* __restrict__ att,
                                float* __restrict__ att_h) {
    const size_t LL4    = (size_t)L_ * L_ / 4;        // float4 elements per (b,h)
    const size_t total4 = (size_t)B_ * LL4;
    for (size_t idx = (size_t)blockIdx.x * blockDim.x + threadIdx.x;
         idx < total4; idx += (size_t)gridDim.x * blockDim.x) {
        size_t b  = idx / LL4;
        size_t rc = idx - b * LL4;
        const v4f* p = (const v4f*)att + b * (size_t)NH_ * LL4 + rc;
        v4f s = {0.f, 0.f, 0.f, 0.f};
        #pragma unroll
        for (int h = 0; h < NH_; ++h) {
            v4f v = __builtin_nontemporal_load(p + (size_t)h * LL4);
            s += v;
        }
        ((v4f*)att_h)[idx] = s * (1.0f / (float)NH_);
    }
}

// ---------------------------------------------------------------------------
// K2: normalized context-attention rows (rows of att_h gathered at mention
// positions; entity rows = mean of 4 mention rows). Each of 256 threads owns
// exactly one float4 of the 1024-wide row; block-reduce for the row sum.
// ---------------------------------------------------------------------------
__global__ void k_ctx_att_rows(const float* __restrict__ att_h,
                               const int* __restrict__ mention_pos,
                               float* __restrict__ norm_att) {
    const int b   = blockIdx.x / CTXROWS_;
    const int row = blockIdx.x % CTXROWS_;
    __shared__ float red[256];

    const int c4 = threadIdx.x;          // float4 index in [0,256)
    v4f v;
    if (row < E_) {
        const int e = row;
        const int p0 = mention_pos[(b * E_ + e) * MPE_ + 0] + 1;
        const int p1 = mention_pos[(b * E_ + e) * MPE_ + 1] + 1;
        const int p2 = mention_pos[(b * E_ + e) * MPE_ + 2] + 1;
        const int p3 = mention_pos[(b * E_ + e) * MPE_ + 3] + 1;
        const v4f* r0 = (const v4f*)(att_h + ((size_t)b * L_ + p0) * L_);
        const v4f* r1 = (const v4f*)(att_h + ((size_t)b * L_ + p1) * L_);
        const v4f* r2 = (const v4f*)(att_h + ((size_t)b * L_ + p2) * L_);
        const v4f* r3 = (const v4f*)(att_h + ((size_t)b * L_ + p3) * L_);
        v = (r0[c4] + r1[c4] + r2[c4] + r3[c4]) * 0.25f;
    } else {
        const int m = row - E_;
        const int p = mention_pos[b * EM_ + m] + 1;   // [B,E,MPE] flat == [B,EM]
        v = ((const v4f*)(att_h + ((size_t)b * L_ + p) * L_))[c4];
    }
    red[threadIdx.x] = v.x + v.y + v.z + v.w;
    __syncthreads();
    for (int off = 128; off > 0; off >>= 1) {
        if ((int)threadIdx.x < off) red[threadIdx.x] += red[threadIdx.x + off];
        __syncthreads();
    }
    const float inv = 1.0f / (red[0] + 1e-5f);
    ((v4f*)(norm_att + ((size_t)b * CTXROWS_ + row) * L_))[c4] = v * inv;
}

// ---------------------------------------------------------------------------
// K3: node bank + type columns + zero padding of ctx type columns.
//   rows 0..31   : entity logsumexp over 4 mention embeddings  (features 0..767)
//   rows 32..159 : mention embedding (gathered seq row)        (features 0..767)
//   all rows     : features 768..787 = type embedding (or 0 for ctx rows)
// ---------------------------------------------------------------------------
__global__ void k_nodes(const float* __restrict__ seq,
                        const float* __restrict__ type_table,
                        const int* __restrict__ mention_pos,
                        float* __restrict__ out) {
    const int b   = blockIdx.x / NROWS_;
    const int row = blockIdx.x % NROWS_;
    float* orow = out + ((size_t)b * NROWS_ + row) * FDIM_;

    if (row < E_) {
        const int e = row;
        const int p0 = mention_pos[(b * E_ + e) * MPE_ + 0] + 1;
        const int p1 = mention_pos[(b * E_ + e) * MPE_ + 1] + 1;
        const int p2 = mention_pos[(b * E_ + e) * MPE_ + 2] + 1;
        const int p3 = mention_pos[(b * E_ + e) * MPE_ + 3] + 1;
        const float* s0 = seq + ((size_t)b * L_ + p0) * H_;
        const float* s1 = seq + ((size_t)b * L_ + p1) * H_;
        const float* s2 = seq + ((size_t)b * L_ + p2) * H_;
        const float* s3 = seq + ((size_t)b * L_ + p3) * H_;
        for (int h = threadIdx.x; h < H_; h += blockDim.x) {
            float x0 = s0[h], x1 = s1[h], x2 = s2[h], x3 = s3[h];
            float m = fmaxf(fmaxf(x0, x1), fmaxf(x2, x3));
            float s = __expf(x0 - m) + __expf(x1 - m) +
                      __expf(x2 - m) + __expf(x3 - m);
            orow[h] = m + __logf(s);
        }
    } else if (row < E_ + EM_) {
        const int m = row - E_;
        const int p = mention_pos[b * EM_ + m] + 1;
        const v4f* sr = (const v4f*)(seq + ((size_t)b * L_ + p) * H_);
        // 768/4 = 192 float4s; orow is 16B aligned (FDIM_*4 = 3152 = 197*16)
        for (int h4 = threadIdx.x; h4 < H_ / 4; h4 += blockDim.x)
            ((v4f*)orow)[h4] = sr[h4];
    }
    // type / padding columns 768..787
    if (threadIdx.x < TD_) {
        float t;
        if      (row < E_)              t = type_table[0 * TD_ + threadIdx.x];
        else if (row < E_ + EM_)        t = type_table[1 * TD_ + threadIdx.x];
        else if (row < NODES_)          t = type_table[2 * TD_ + threadIdx.x];
        else                            t = 0.0f;   // ctx rows zero-padded
        orow[H_ + threadIdx.x] = t;
    }
}

// ---------------------------------------------------------------------------
// K4: link span pooling. One 256-thread block per (b,k).
//   colw[j] = sum_{r in span} att_h[b, s+r, s+j]     (span length <= 31)
//   link_rep[h] = sum_j colw[j] * seq[b, s+j, h] / len
// ---------------------------------------------------------------------------
__global__ void k_links(const float* __restrict__ seq,
                        const float* __restrict__ att_h,
                        const int* __restrict__ link_start,
                        const int* __restrict__ link_len,
                        float* __restrict__ out) {
    const int b = blockIdx.x / K_;
    const int k = blockIdx.x % K_;
    const int s   = link_start[b * K_ + k] + 1;   // + OFFSET
    const int len = link_len[b * K_ + k] + 1;     // span length in [1,31]
    __shared__ float colw[32];

    if ((int)threadIdx.x < len) {
        const int c = s + threadIdx.x;
        float acc = 0.f;
        for (int r = 0; r < len; ++r)
            acc += att_h[((size_t)b * L_ + s + r) * L_ + c];
        colw[threadIdx.x] = acc;
    }
    __syncthreads();

    const float invlen = 1.0f / (float)len;
    float* orow = out + ((size_t)b * NROWS_ + (E_ + EM_) + k) * FDIM_;
    for (int h = threadIdx.x; h < H_; h += blockDim.x) {
        float acc = 0.f;
        for (int j = 0; j < len; ++j)
            acc += colw[j] * seq[((size_t)b * L_ + s + j) * H_ + h];
        orow[h] = acc * invlen;
    }
}

// ---------------------------------------------------------------------------
// K5: ctx GEMM via V_WMMA_F32_16X16X4_F32 (exact f32 matrix path).
//   ctx[b,160,768] = norm_att[b,160,1024] @ seq[b,1024,768]
// One wave per 16x32 output tile: A (16x4) loaded once per k-step, two B
// sub-tiles -> two WMMAs, halving A traffic vs a 16x16 tile.
// A 16x4 f32 layout (ISA 7.12.2): lanes 0-15 row=lane, VGPR0/1 = K0/K1;
//                                 lanes 16-31 row=lane-16, VGPR0/1 = K2/K3.
// C/D: VGPR i -> row i (lanes 0-15) / row i+8 (lanes 16-31).
// Tiles: B * (160/16) * (768/32) = 960 waves = 120 blocks x 8 waves.
// EXEC all-1s (exact tile coverage) as WMMA requires.
// ---------------------------------------------------------------------------
__global__ void k_ctx_gemm_wmma(const float* __restrict__ norm_att,
                                const float* __restrict__ seq,
                                float* __restrict__ out) {
    const int MT = CTXROWS_ / 16;   // 10
    const int NT = H_ / 32;         // 24
    const int wave = (int)((blockIdx.x * blockDim.x + threadIdx.x) >> 5);
    const int lane = threadIdx.x & 31;
    const int b   = wave / (MT * NT);
    const int rem = wave % (MT * NT);
    const int m0 = (rem / NT) * 16;
    const int n0 = (rem % NT) * 32;
    const int half = lane >> 4;     // 0: K0/K1, 1: K2/K3
    const int l16  = lane & 15;

    const float* Arow  = norm_att + ((size_t)b * CTXROWS_ + m0 + l16) * L_;
    const float* Bcol0 = seq + (size_t)b * L_ * H_ + n0 + l16;
    const float* Bcol1 = Bcol0 + 16;

    v8f acc0 = {0.f, 0.f, 0.f, 0.f, 0.f, 0.f, 0.f, 0.f};
    v8f acc1 = {0.f, 0.f, 0.f, 0.f, 0.f, 0.f, 0.f, 0.f};

    #pragma unroll 4
    for (int k0 = 0; k0 < L_; k0 += 4) {
        const int ka = k0 + half * 2;
        v2f a = *(const v2f*)(Arow + ka);          // A[row, ka], A[row, ka+1]
        v2f b0, b1;
        b0.x = Bcol0[(size_t)ka * H_];             // B[ka,   col]
        b0.y = Bcol0[(size_t)(ka + 1) * H_];       // B[ka+1, col]
        b1.x = Bcol1[(size_t)ka * H_];
        b1.y = Bcol1[(size_t)(ka + 1) * H_];
        acc0 = __builtin_amdgcn_wmma_f32_16x16x4_f32(
            false, a, false, b0, (short)0, acc0, false, false);
        acc1 = __builtin_amdgcn_wmma_f32_16x16x4_f32(
            false, a, false, b1, (short)0, acc1, false, false);
    }

    // D layout: VGPR i -> row m0 + i (lanes 0-15) / m0 + 8 + i (lanes 16-31)
    float* obase = out + ((size_t)b * NROWS_ + NODES_ + m0 + half * 8) * FDIM_
                 + n0 + l16;
    #pragma unroll
    for (int i = 0; i < 8; ++i) {
        obase[(size_t)i * FDIM_]      = acc0[i];
        obase[(size_t)i * FDIM_ + 16] = acc1[i];
    }
}

// ---------------------------------------------------------------------------
extern "C" void kernel_launch(void* const* d_in, const int* in_sizes, int n_in,
                              void* d_out, int out_size, void* d_ws, size_t ws_size,
                              hipStream_t stream) {
    const float* seq         = (const float*)d_in[0];  // [B,L,H]
    const float* att         = (const float*)d_in[1];  // [B,NH,L,L]
    const float* type_table  = (const float*)d_in[2];  // [3,TD]
    const int*   mention_pos = (const int*)d_in[3];    // [B,E,MPE]
    const int*   link_start  = (const int*)d_in[4];    // [B,K]
    const int*   link_len    = (const int*)d_in[5];    // [B,K]
    float* out = (float*)d_out;                        // [B,336,788]

    float* att_h    = (float*)d_ws;                      // B*L*L floats (16 MiB)
    float* norm_att = att_h + (size_t)B_ * L_ * L_;      // B*160*L floats

    k_att_head_mean<<<2048, 256, 0, stream>>>(att, att_h);
    k_ctx_att_rows<<<B_ * CTXROWS_, 256, 0, stream>>>(att_h, mention_pos, norm_att);
    k_nodes<<<B_ * NROWS_, 256, 0, stream>>>(seq, type_table, mention_pos, out);
    k_links<<<B_ * K_, 256, 0, stream>>>(seq, att_h, link_start, link_len, out);
    k_ctx_gemm_wmma<<<120, 256, 0, stream>>>(norm_att, seq, out);
}